// GaussianRenderer_3667902071213
// MI455X (gfx1250) — compile-verified
//
#include <hip/hip_runtime.h>
#include <hip/hip_bf16.h>

#define NG    2048
#define IMGH  128
#define IMGW  128
#define FXc   160.0f
#define FYc   160.0f
#define CXc   64.0f
#define CYc   64.0f
#define EPSc  1e-6f

typedef float v2f __attribute__((ext_vector_type(2)));
typedef float v8f __attribute__((ext_vector_type(8)));

// ---------------------------------------------------------------------------
// Kernel 1: per-gaussian projection, 2D covariance inverse, color/alpha pack.
// pre[i*9]  = {mx, my, i00, i01, i11, cr, cg, cb, alpha};  depth[i] = z
// ---------------------------------------------------------------------------
__global__ __launch_bounds__(256) void gs_preprocess(
    const float* __restrict__ mean, const float* __restrict__ qvec,
    const float* __restrict__ svec, const float* __restrict__ color,
    const float* __restrict__ alpha, const float* __restrict__ c2w,
    float* __restrict__ pre, float* __restrict__ depth)
{
    int i = blockIdx.x * blockDim.x + threadIdx.x;
    if (i >= NG) return;

    // Wm = c2w[:, :3]^T, d = -c2w[:, 3]
    float Wm[3][3];
#pragma unroll
    for (int r = 0; r < 3; ++r)
#pragma unroll
        for (int c = 0; c < 3; ++c)
            Wm[r][c] = c2w[c * 4 + r];
    float d0 = -c2w[3], d1 = -c2w[7], d2 = -c2w[11];

    float m0 = mean[i * 3 + 0] + d0;
    float m1 = mean[i * 3 + 1] + d1;
    float m2 = mean[i * 3 + 2] + d2;
    float pm0 = Wm[0][0] * m0 + Wm[0][1] * m1 + Wm[0][2] * m2;
    float pm1 = Wm[1][0] * m0 + Wm[1][1] * m1 + Wm[1][2] * m2;
    float pm2 = Wm[2][0] * m0 + Wm[2][1] * m1 + Wm[2][2] * m2;

    // normalized quaternion -> rotation, scaled columns
    float qw_ = qvec[i * 4 + 0], qx_ = qvec[i * 4 + 1];
    float qy_ = qvec[i * 4 + 2], qz_ = qvec[i * 4 + 3];
    float qinv = 1.0f / sqrtf(qw_ * qw_ + qx_ * qx_ + qy_ * qy_ + qz_ * qz_);
    float w = qw_ * qinv, x = qx_ * qinv, y = qy_ * qinv, z = qz_ * qinv;

    float R00 = 1.0f - 2.0f * (y * y + z * z);
    float R01 = 2.0f * (x * y - w * z);
    float R02 = 2.0f * (x * z + w * y);
    float R10 = 2.0f * (x * y + w * z);
    float R11 = 1.0f - 2.0f * (x * x + z * z);
    float R12 = 2.0f * (y * z - w * x);
    float R20 = 2.0f * (x * z - w * y);
    float R21 = 2.0f * (y * z + w * x);
    float R22 = 1.0f - 2.0f * (x * x + y * y);

    float s0 = svec[i * 3 + 0], s1 = svec[i * 3 + 1], s2 = svec[i * 3 + 2];
    float M00 = R00 * s0, M01 = R01 * s1, M02 = R02 * s2;
    float M10 = R10 * s0, M11 = R11 * s1, M12 = R12 * s2;
    float M20 = R20 * s0, M21 = R21 * s1, M22 = R22 * s2;

    float sig00 = M00 * M00 + M01 * M01 + M02 * M02;
    float sig01 = M00 * M10 + M01 * M11 + M02 * M12;
    float sig02 = M00 * M20 + M01 * M21 + M02 * M22;
    float sig11 = M10 * M10 + M11 * M11 + M12 * M12;
    float sig12 = M10 * M20 + M11 * M21 + M12 * M22;
    float sig22 = M20 * M20 + M21 * M21 + M22 * M22;

    float zc = pm2;
    float invz = 1.0f / zc;
    float mx = pm0 * invz, my = pm1 * invz;

    // rows 0,1 of J @ Wm (row 2 never reaches the 2x2 block)
    float JW0[3], JW1[3];
#pragma unroll
    for (int j = 0; j < 3; ++j) {
        JW0[j] = (Wm[0][j] - mx * Wm[2][j]) * invz;
        JW1[j] = (Wm[1][j] - my * Wm[2][j]) * invz;
    }
    float t0 = sig00 * JW0[0] + sig01 * JW0[1] + sig02 * JW0[2];
    float t1 = sig01 * JW0[0] + sig11 * JW0[1] + sig12 * JW0[2];
    float t2 = sig02 * JW0[0] + sig12 * JW0[1] + sig22 * JW0[2];
    float u0 = sig00 * JW1[0] + sig01 * JW1[1] + sig02 * JW1[2];
    float u1 = sig01 * JW1[0] + sig11 * JW1[1] + sig12 * JW1[2];
    float u2 = sig02 * JW1[0] + sig12 * JW1[1] + sig22 * JW1[2];
    float cov00 = JW0[0] * t0 + JW0[1] * t1 + JW0[2] * t2;
    float cov01 = JW1[0] * t0 + JW1[1] * t1 + JW1[2] * t2;
    float cov11 = JW1[0] * u0 + JW1[1] * u1 + JW1[2] * u2;

    float a_ = cov00 + EPSc;
    float b_ = cov01;
    float dd = cov11 + EPSc;
    float det = a_ * dd - b_ * b_;
    float idet = 1.0f / det;
    float i00 = dd * idet;
    float i01 = -b_ * idet;
    float i11 = a_ * idet;

    float* p = pre + i * 9;
    p[0] = mx;  p[1] = my;
    p[2] = i00; p[3] = i01; p[4] = i11;
    p[5] = color[i * 3 + 0];
    p[6] = color[i * 3 + 1];
    p[7] = color[i * 3 + 2];
    p[8] = alpha[i];
    depth[i] = zc;
}

// ---------------------------------------------------------------------------
// Kernel 2: bitonic sort by depth (single block, LDS) + build sorted records
// rec[i*10] = {B0..B5 (bilinear-form coeffs), cr, cg, cb, alpha}
// power(p,n) = B0*px^2 + B1*py^2 + B2*px*py + B3*px + B4*py + B5
// ---------------------------------------------------------------------------
__global__ __launch_bounds__(1024) void gs_sort_build(
    const float* __restrict__ depth, const float* __restrict__ pre,
    float* __restrict__ rec)
{
    __shared__ float sk[NG];
    __shared__ int   sv[NG];
    int tid = threadIdx.x;
    for (int i = tid; i < NG; i += 1024) { sk[i] = depth[i]; sv[i] = i; }
    __syncthreads();

    for (int k = 2; k <= NG; k <<= 1) {
        for (int j = k >> 1; j > 0; j >>= 1) {
            for (int i = tid; i < NG; i += 1024) {
                int ixj = i ^ j;
                if (ixj > i) {
                    bool asc = ((i & k) == 0);
                    float a = sk[i], b = sk[ixj];
                    if ((a > b) == asc) {
                        sk[i] = b; sk[ixj] = a;
                        int t = sv[i]; sv[i] = sv[ixj]; sv[ixj] = t;
                    }
                }
            }
            __syncthreads();
        }
    }

    for (int i = tid; i < NG; i += 1024) {
        int s = sv[i];
        const float* p = pre + s * 9;
        float mx = p[0], my = p[1], i00 = p[2], i01 = p[3], i11 = p[4];
        float* r = rec + i * 10;
        r[0] = -0.5f * i00;
        r[1] = -0.5f * i11;
        r[2] = -i01;
        r[3] = i00 * mx + i01 * my;
        r[4] = i11 * my + i01 * mx;
        r[5] = -0.5f * (i00 * mx * mx + 2.0f * i01 * mx * my + i11 * my * my);
        r[6] = p[5]; r[7] = p[6]; r[8] = p[7]; r[9] = p[8];
    }
}

// ---------------------------------------------------------------------------
// Kernel 3: render. Each wave owns 16 pixels; per 16-gaussian tile two
// V_WMMA_F32_16X16X4_F32 produce the 16x16 fp32 power tile (K=8 features,
// split K=4+4). Transmittance scan = width-16 shuffle prefix product.
// A layout (16x4 fp32): lanes 0-15 -> {K0,K1}, lanes 16-31 -> {K2,K3}.
// B layout (4x16 fp32): same K split across lane halves, N = lane%16.
// D layout: VGPR r -> pixel {r | r+8 by lane half}, N = lane%16.
// ---------------------------------------------------------------------------
__global__ __launch_bounds__(256) void gs_render(
    const float* __restrict__ rec, float* __restrict__ out)
{
    const int lane = threadIdx.x & 31;
    const int wave = threadIdx.x >> 5;
    const int base = (blockIdx.x * 8 + wave) * 16;
    const bool hi  = lane >= 16;
    const int  l16 = lane & 15;

    // pixel features for A-matrix row M = l16
    int pA = base + l16;
    float xA = (float)(pA & (IMGW - 1));
    float yA = (float)(pA >> 7);
    float px = (xA + 0.5f) / FXc - CXc / FXc;
    float py = (yA + 0.5f) / FYc - CYc / FYc;

    v2f a1, a2;
    a1.x = hi ? px * py : px * px;   // K0=px^2 | K2=px*py
    a1.y = hi ? px      : py * py;   // K1=py^2 | K3=px
    a2.x = hi ? 0.0f    : py;        // K4=py   | K6=0
    a2.y = hi ? 0.0f    : 1.0f;      // K5=1    | K7=0

    float T[8], aR[8], aG[8], aB[8];
#pragma unroll
    for (int r = 0; r < 8; ++r) { T[r] = 1.0f; aR[r] = aG[r] = aB[r] = 0.0f; }

    for (int t = 0; t < NG / 16; ++t) {
        const float* rp = rec + (t * 16 + l16) * 10;
        float B0 = rp[0], B1 = rp[1], B2 = rp[2];
        float B3 = rp[3], B4 = rp[4], B5 = rp[5];
        float cr = rp[6], cg = rp[7], cb = rp[8], al = rp[9];

        v2f b1, b2;
        b1.x = hi ? B2 : B0;
        b1.y = hi ? B3 : B1;
        b2.x = hi ? 0.0f : B4;
        b2.y = hi ? 0.0f : B5;

        v8f acc = {};
        acc = __builtin_amdgcn_wmma_f32_16x16x4_f32(false, a1, false, b1,
                                                    (short)0, acc, false, false);
        acc = __builtin_amdgcn_wmma_f32_16x16x4_f32(false, a2, false, b2,
                                                    (short)0, acc, false, false);

#pragma unroll
        for (int r = 0; r < 8; ++r) {
            float w = al * __expf(acc[r]);
            w = fminf(w, 0.999f);
            w = (w > 0.001f) ? w : 0.0f;
            float om = 1.0f - w;

            // inclusive prefix product over the 16-lane group (depth order)
            float p = om;
#pragma unroll
            for (int off = 1; off < 16; off <<= 1) {
                float tsh = __shfl_up(p, off, 16);
                if (l16 >= off) p *= tsh;
            }
            float excl = __shfl_up(p, 1, 16);
            if (l16 == 0) excl = 1.0f;

            float cw = w * T[r] * excl;
            aR[r] += cw * cr;
            aG[r] += cw * cg;
            aB[r] += cw * cb;

            float tot = __shfl(p, 15, 16);  // product of all 16 factors
            T[r] *= tot;
        }

        // wave-uniform early out keeps EXEC all-ones for the WMMAs
        float tmax = T[0];
#pragma unroll
        for (int r = 1; r < 8; ++r) tmax = fmaxf(tmax, T[r]);
        if (!__any(tmax > 1e-7f)) break;
    }

#pragma unroll
    for (int r = 0; r < 8; ++r) {
#pragma unroll
        for (int off = 8; off > 0; off >>= 1) {
            aR[r] += __shfl_xor(aR[r], off, 16);
            aG[r] += __shfl_xor(aG[r], off, 16);
            aB[r] += __shfl_xor(aB[r], off, 16);
        }
        if (l16 == 0) {
            int pid = base + r + (hi ? 8 : 0);
            out[pid * 3 + 0] = aR[r];
            out[pid * 3 + 1] = aG[r];
            out[pid * 3 + 2] = aB[r];
        }
    }
}

// ---------------------------------------------------------------------------
extern "C" void kernel_launch(void* const* d_in, const int* in_sizes, int n_in,
                              void* d_out, int out_size, void* d_ws, size_t ws_size,
                              hipStream_t stream)
{
    (void)in_sizes; (void)n_in; (void)out_size; (void)ws_size;
    const float* mean  = (const float*)d_in[0];
    const float* qvec  = (const float*)d_in[1];
    const float* svec  = (const float*)d_in[2];
    const float* color = (const float*)d_in[3];
    const float* alpha = (const float*)d_in[4];
    const float* c2w   = (const float*)d_in[5];

    float* ws    = (float*)d_ws;
    float* pre   = ws;             // NG*9 floats
    float* depth = ws + NG * 9;    // NG floats
    float* rec   = ws + NG * 10;   // NG*10 floats  (total 160 KB)
    float* out   = (float*)d_out;

    hipLaunchKernelGGL(gs_preprocess, dim3(NG / 256), dim3(256), 0, stream,
                       mean, qvec, svec, color, alpha, c2w, pre, depth);
    hipLaunchKernelGGL(gs_sort_build, dim3(1), dim3(1024), 0, stream,
                       depth, pre, rec);
    hipLaunchKernelGGL(gs_render, dim3((IMGH * IMGW) / 128), dim3(256), 0, stream,
                       rec, out);
}